// SharedSharedLoss_73297911873997
// MI455X (gfx1250) — compile-verified
//
#include <hip/hip_runtime.h>

// ---------------------------------------------------------------------------
// Pairwise mean off-diagonal squared-L2 distance over A[4096, 512] fp32.
//   answer = (2N*sum_i ||a_i||^2 - 2*sum_{ij} gram_ij) / (C * N * (N-1))
// Gram sum via fp32 WMMA (V_WMMA_F32_16X16X4_F32). Symmetry: only the lower
// triangle of 32x32 super-tiles is computed; off-diagonal tile sums count
// twice. Each 16x16 accumulator tile is collapsed to a scalar in-register —
// the N x N Gram (64 MB) never touches memory. Input (8 MB) is L2-resident.
// ---------------------------------------------------------------------------

typedef __attribute__((ext_vector_type(2))) float v2f;
typedef __attribute__((ext_vector_type(8))) float v8f;

static constexpr int kN  = 4096;
static constexpr int kC  = 512;
static constexpr int kST = 128;                       // 32-row super-tiles per dim
static constexpr int kTri = kST * (kST + 1) / 2;      // 8256 lower-tri super-tiles

__global__ void ssl_init_ws(float* ws) {
    if (threadIdx.x < 2) ws[threadIdx.x] = 0.0f;
}

// --- sum of squares of all elements (memory-bound O(N*C) reduction) --------
__global__ void __launch_bounds__(256) ssl_sumsq(const float* __restrict__ A,
                                                 float* __restrict__ ws) {
    const size_t total  = (size_t)kN * (size_t)kC;
    const size_t stride = (size_t)gridDim.x * blockDim.x;
    float acc = 0.0f;
    for (size_t i = (size_t)blockIdx.x * blockDim.x + threadIdx.x; i < total; i += stride) {
        float v = A[i];
        acc = fmaf(v, v, acc);
    }
    for (int off = 16; off > 0; off >>= 1) acc += __shfl_xor(acc, off, 32);
    __shared__ float red[8];
    const int lane = threadIdx.x & 31;
    const int wv   = threadIdx.x >> 5;
    if (lane == 0) red[wv] = acc;
    __syncthreads();
    if (threadIdx.x == 0) {
        float s = 0.0f;
        for (int j = 0; j < 8; ++j) s += red[j];
        atomicAdd(&ws[0], s);
    }
}

// --- Gram total via fp32 WMMA, lower-triangular super-tiles ----------------
// One wave per 32x32 super-tile (2x2 grid of 16x16 WMMA tiles, K = 512 in
// steps of 4). A-fragment (16x4 fp32) per-lane layout:
//   lane = 16*half + r :  v0 = A[M=r][k+2*half], v1 = A[M=r][k+2*half+1]
// B = A^T, so the B-fragment uses the same load pattern from the column
// block's rows. Tile sum weighted x2 off-diagonal (Gram symmetry).
__global__ void __launch_bounds__(256) ssl_gram(const float* __restrict__ A,
                                                float* __restrict__ ws) {
    const int lane = threadIdx.x & 31;
    const int wv   = threadIdx.x >> 5;
    const int half = lane >> 4;   // 0 or 1
    const int r    = lane & 15;

    const int t = blockIdx.x * 8 + wv;          // 0 .. kTri-1 (wave-uniform)
    if (t >= kTri) return;

    // Decode lower-triangle (i >= j): t = i*(i+1)/2 + j
    int i = (int)((__builtin_sqrtf(8.0f * (float)t + 1.0f) - 1.0f) * 0.5f);
    while ((i + 1) * (i + 2) / 2 <= t) ++i;     // fix float rounding
    while (i * (i + 1) / 2 > t) --i;
    const int j  = t - i * (i + 1) / 2;
    const int sm = i << 5;                       // row base of 32x32 super-tile
    const int sn = j << 5;                       // col base

    const float* __restrict__ rowA0 = A + (size_t)(sm + r) * kC;
    const float* __restrict__ rowA1 = rowA0 + (size_t)16 * kC;
    const float* __restrict__ rowB0 = A + (size_t)(sn + r) * kC;
    const float* __restrict__ rowB1 = rowB0 + (size_t)16 * kC;

    v8f acc00 = {}, acc01 = {}, acc10 = {}, acc11 = {};

#pragma unroll 4
    for (int k = 0; k < kC; k += 4) {
        const int o = k + 2 * half;
        v2f a0 = *(const v2f*)(rowA0 + o);
        v2f a1 = *(const v2f*)(rowA1 + o);
        v2f b0 = *(const v2f*)(rowB0 + o);
        v2f b1 = *(const v2f*)(rowB1 + o);
        acc00 = __builtin_amdgcn_wmma_f32_16x16x4_f32(false, a0, false, b0,
                                                      (short)0, acc00, false, false);
        acc01 = __builtin_amdgcn_wmma_f32_16x16x4_f32(false, a0, false, b1,
                                                      (short)0, acc01, false, false);
        acc10 = __builtin_amdgcn_wmma_f32_16x16x4_f32(false, a1, false, b0,
                                                      (short)0, acc10, false, false);
        acc11 = __builtin_amdgcn_wmma_f32_16x16x4_f32(false, a1, false, b1,
                                                      (short)0, acc11, false, false);
    }

    float s = 0.0f;
#pragma unroll
    for (int q = 0; q < 8; ++q) s += acc00[q] + acc01[q] + acc10[q] + acc11[q];
    s *= (i == j) ? 1.0f : 2.0f;                // symmetry weight
    for (int off = 16; off > 0; off >>= 1) s += __shfl_xor(s, off, 32);
    if (lane == 0) atomicAdd(&ws[1], s);
}

__global__ void ssl_finalize(const float* __restrict__ ws, float* __restrict__ out) {
    const double sumsq = (double)ws[0];
    const double gsum  = (double)ws[1];
    const double total = (2.0 * (double)kN * sumsq - 2.0 * gsum) / (double)kC;
    out[0] = (float)(total / ((double)kN * (double)(kN - 1)));
}

extern "C" void kernel_launch(void* const* d_in, const int* in_sizes, int n_in,
                              void* d_out, int out_size, void* d_ws, size_t ws_size,
                              hipStream_t stream) {
    (void)in_sizes; (void)n_in; (void)out_size; (void)ws_size;
    const float* A   = (const float*)d_in[0];
    float*       ws  = (float*)d_ws;    // ws[0] = sum of squares, ws[1] = gram total
    float*       out = (float*)d_out;

    ssl_init_ws<<<1, 64, 0, stream>>>(ws);
    ssl_sumsq<<<256, 256, 0, stream>>>(A, ws);
    // 8256 lower-triangular super-tiles, 8 waves per block -> 1032 blocks
    ssl_gram<<<(kTri + 7) / 8, 256, 0, stream>>>(A, ws);
    ssl_finalize<<<1, 1, 0, stream>>>(ws, out);
}